// SpikeFP32Divider_48988396978588
// MI455X (gfx1250) — compile-verified
//
#include <hip/hip_runtime.h>
#include <stdint.h>

// SpikeFP32Divider: bitwise spike FP32 divide.
// Per element: pack 32 spike floats -> u32 (MSB first), IEEE f32 divide, unpack.
// Memory-bound streaming kernel (768 MB @ 23.3 TB/s ~= 33 us floor on MI455X).

typedef float f32x4 __attribute__((ext_vector_type(4)));

__device__ __forceinline__ uint32_t pack_row_nt(const f32x4* __restrict__ p) {
    uint32_t u = 0u;
#pragma unroll
    for (int w = 0; w < 8; ++w) {
        // 128-bit non-temporal load: global_load_b128 th:TH_LOAD_NT
        f32x4 v = __builtin_nontemporal_load(p + w);
        const int j = w * 4;  // memory bit index of v.x; value bit = 31 - j
        u |= (v.x != 0.0f ? 1u : 0u) << (31 - j);
        u |= (v.y != 0.0f ? 1u : 0u) << (30 - j);
        u |= (v.z != 0.0f ? 1u : 0u) << (29 - j);
        u |= (v.w != 0.0f ? 1u : 0u) << (28 - j);
    }
    return u;
}

__device__ __forceinline__ void unpack_row_nt(uint32_t q, f32x4* __restrict__ o) {
#pragma unroll
    for (int w = 0; w < 8; ++w) {
        const int j = w * 4;
        f32x4 v;
        v.x = (float)((q >> (31 - j)) & 1u);
        v.y = (float)((q >> (30 - j)) & 1u);
        v.z = (float)((q >> (29 - j)) & 1u);
        v.w = (float)((q >> (28 - j)) & 1u);
        // 128-bit non-temporal store: global_store_b128 th:TH_STORE_NT
        __builtin_nontemporal_store(v, o + w);
    }
}

__global__ void __launch_bounds__(256)
spike_fp32_div_kernel(const float* __restrict__ A,
                      const float* __restrict__ B,
                      float* __restrict__ O,
                      int n_elems) {
    const int stride = (int)(gridDim.x * blockDim.x);
    for (int e = (int)(blockIdx.x * blockDim.x + threadIdx.x); e < n_elems; e += stride) {
        const size_t base = (size_t)e * 32u;

        // Prefetch next grid-stride iteration's rows (global_prefetch_b8):
        // hides HBM latency behind the ~2 rows of pack/divide/unpack VALU work.
        const int en = e + stride;
        if (en < n_elems) {
            const size_t nb = (size_t)en * 32u;
            __builtin_prefetch(A + nb, 0, 0);
            __builtin_prefetch(B + nb, 0, 0);
        }

        const uint32_t ua = pack_row_nt((const f32x4*)(A + base));
        const uint32_t ub = pack_row_nt((const f32x4*)(B + base));

        // Correctly-rounded IEEE f32 divide (v_div_scale/v_div_fmas/v_div_fixup,
        // RNE, denorms allowed) — bit-exact vs the reference's float32 divide,
        // including NaN/Inf/zero/subnormal special cases.
        const float q = __uint_as_float(ua) / __uint_as_float(ub);

        unpack_row_nt(__float_as_uint(q), (f32x4*)(O + base));
    }
}

extern "C" void kernel_launch(void* const* d_in, const int* in_sizes, int n_in,
                              void* d_out, int out_size, void* d_ws, size_t ws_size,
                              hipStream_t stream) {
    const float* A = (const float*)d_in[0];   // [N, 32] spike floats
    const float* B = (const float*)d_in[1];   // [N, 32] spike floats
    float* O = (float*)d_out;                 // [N, 32] spike floats

    const int n_elems = in_sizes[0] / 32;     // N = 2,097,152

    // wave32: 256 threads = 8 waves per block. ~4 elements per thread via
    // grid-stride keeps ~16K waves in flight (saturates HBM) and gives the
    // prefetch a full iteration of latency-hiding distance.
    const int block = 256;
    const int elems_per_thread = 4;
    long long threads = ((long long)n_elems + elems_per_thread - 1) / elems_per_thread;
    int grid = (int)((threads + block - 1) / block);
    if (grid < 1) grid = 1;

    spike_fp32_div_kernel<<<grid, block, 0, stream>>>(A, B, O, n_elems);
}